// DiffKGBase_12378095747627
// MI455X (gfx1250) — compile-verified
//
#include <hip/hip_runtime.h>
#include <stdint.h>

// ---------------------------------------------------------------------------
// DiffKG multi-hop sparse walk, MI455X (gfx1250, wave32).
//   per hop: score = e[b,heads]*r[b,rels]; walked = segsum(score -> tails);
//            walked /= (rowsum + 1e-6)
// Layout: entity/walked arrays transposed [E,16] so the 16 lanes of a
// half-wave cover the batch dim -> every gather and every scatter-atomic is
// one coalesced 64B transaction. Index streams staged into LDS with CDNA5
// async global->LDS loads, double buffered; outputs stored non-temporally.
// ---------------------------------------------------------------------------

static constexpr int BATCH = 16;
static constexpr int HOPS  = 3;
static constexpr int TILE_EDGES      = 256;
static constexpr int SCATTER_BLOCKS  = 2048;
static constexpr int SCATTER_THREADS = 256;

#define USE_ASYNC_LDS 1

// ---- CDNA5 async global->LDS helpers (CDNA5 ISA) ----
__device__ __forceinline__ void async_g2l_b32(const void* gaddr, void* lds_addr) {
#if USE_ASYNC_LDS
  // Flat LDS addresses carry the LDS byte offset in the low 32 bits
  // (ISA §10.2 aperture rules), so truncation gives the AS3 offset.
  unsigned lds_off = (unsigned)(uintptr_t)lds_addr;
  unsigned long long ga = (unsigned long long)(uintptr_t)gaddr;
  asm volatile("global_load_async_to_lds_b32 %0, %1, off"
               :: "v"(lds_off), "v"(ga)
               : "memory");
#else
  *(int*)lds_addr = *(const int*)gaddr;
#endif
}

__device__ __forceinline__ void wait_async_g2l() {
#if USE_ASYNC_LDS
  asm volatile("s_wait_asynccnt 0x0" ::: "memory");
#endif
}

// Native device-scope f32 atomic add, no-return form (global_atomic_add_f32,
// tracked on STOREcnt). Avoids any CAS-loop lowering of generic atomicAdd.
__device__ __forceinline__ void fatomic_add(float* p, float v) {
  (void)__hip_atomic_fetch_add(p, v, __ATOMIC_RELAXED, __HIP_MEMORY_SCOPE_AGENT);
}

// ---------------------------------------------------------------------------
// out[:,0,:] = init_ent   (streaming copy, NT both sides)
__global__ void k_copy_hop0(const float* __restrict__ init_ent,
                            float* __restrict__ out, int E, long long outB) {
  int b = blockIdx.y;
  for (long long e = (long long)blockIdx.x * blockDim.x + threadIdx.x; e < E;
       e += (long long)gridDim.x * blockDim.x) {
    float v = __builtin_nontemporal_load(&init_ent[(long long)b * E + e]);
    __builtin_nontemporal_store(v, &out[(long long)b * outB + e]);
  }
}

// init_ent [B,E] -> ent_t [E,16]
__global__ void k_transpose_in(const float* __restrict__ init_ent,
                               float* __restrict__ ent_t, int E) {
  long long n = (long long)E * BATCH;
  for (long long i = (long long)blockIdx.x * blockDim.x + threadIdx.x; i < n;
       i += (long long)gridDim.x * blockDim.x) {
    long long e = i >> 4;
    int       b = (int)(i & (BATCH - 1));
    ent_t[i] = init_ent[(long long)b * E + e];
  }
}

// rels_seq[:,hop,:] [B,R] -> r_t [R,16]
__global__ void k_prep_r(const float* __restrict__ rels_seq,
                         float* __restrict__ r_t, int hop, int R) {
  int i = blockIdx.x * blockDim.x + threadIdx.x;
  if (i < R * BATCH) {
    int rl = i >> 4, b = i & (BATCH - 1);
    r_t[i] = rels_seq[((long long)b * HOPS + hop) * R + rl];
  }
}

__global__ void k_zero(float* __restrict__ p, long long n) {
  for (long long i = (long long)blockIdx.x * blockDim.x + threadIdx.x; i < n;
       i += (long long)gridDim.x * blockDim.x)
    p[i] = 0.0f;
}

// zero a strided [B,E] slice (fallback path, lives inside d_out)
__global__ void k_zero_slice(float* __restrict__ base, long long sb, int E) {
  int b = blockIdx.y;
  for (long long e = (long long)blockIdx.x * blockDim.x + threadIdx.x; e < E;
       e += (long long)gridDim.x * blockDim.x)
    base[(long long)b * sb + e] = 0.0f;
}

// ---------------------------------------------------------------------------
// Gather-multiply-scatter over edges. 16 lanes <-> 16 batches, 2 edges/wave.
// Index triples staged to LDS with async loads, double buffered.
__global__ __launch_bounds__(SCATTER_THREADS)
void k_scatter(const float* __restrict__ ent, long long es_e, long long es_b,
               const float* __restrict__ rmat, long long rs_r, long long rs_b,
               float* __restrict__ walked, long long ws_e, long long ws_b,
               const int* __restrict__ heads, const int* __restrict__ rels,
               const int* __restrict__ tails, int T) {
  __shared__ int sh[2][3][TILE_EDGES];
  const int tid    = threadIdx.x;
  const int lane_b = tid & (BATCH - 1);
  const int sub    = tid >> 4;                  // edge slot within iteration
  const int epi    = blockDim.x / BATCH;        // edges per inner iteration

  const long long nTiles = ((long long)T + TILE_EDGES - 1) / TILE_EDGES;
  long long tile = blockIdx.x;
  int buf = 0;

  if (tile < nTiles) {
    long long t0 = tile * TILE_EDGES;
    if (t0 + tid < T) {
      async_g2l_b32(heads + t0 + tid, &sh[buf][0][tid]);
      async_g2l_b32(rels  + t0 + tid, &sh[buf][1][tid]);
      async_g2l_b32(tails + t0 + tid, &sh[buf][2][tid]);
    }
  }

  for (; tile < nTiles; tile += gridDim.x) {
    wait_async_g2l();          // this wave's staged loads for `buf` landed
    __syncthreads();           // all waves' loads landed

    // issue next tile into the other buffer, prefetch the one after that
    long long ntile = tile + gridDim.x;
    if (ntile < nTiles) {
      long long t0n = ntile * TILE_EDGES;
      if (t0n + tid < T) {
        async_g2l_b32(heads + t0n + tid, &sh[buf ^ 1][0][tid]);
        async_g2l_b32(rels  + t0n + tid, &sh[buf ^ 1][1][tid]);
        async_g2l_b32(tails + t0n + tid, &sh[buf ^ 1][2][tid]);
      }
      long long t0nn = t0n + (long long)gridDim.x * TILE_EDGES;
      if (t0nn + tid < T) {    // pull the tile after that toward L2
        __builtin_prefetch(heads + t0nn + tid, 0, 0);
        __builtin_prefetch(rels  + t0nn + tid, 0, 0);
        __builtin_prefetch(tails + t0nn + tid, 0, 0);
      }
    }

    long long t0  = tile * TILE_EDGES;
    long long rem = (long long)T - t0;
    int cnt = rem < (long long)TILE_EDGES ? (int)rem : TILE_EDGES;

    for (int e = sub; e < cnt; e += epi) {
      int h  = sh[buf][0][e];
      int rl = sh[buf][1][e];
      int tl = sh[buf][2][e];
      float ev = ent [(long long)h  * es_e + (long long)lane_b * es_b];
      float rv = rmat[(long long)rl * rs_r + (long long)lane_b * rs_b];
      fatomic_add(&walked[(long long)tl * ws_e + (long long)lane_b * ws_b],
                  ev * rv);
    }
    __syncthreads();           // everyone done reading `buf` before reuse
    buf ^= 1;
  }
}

// ---------------------------------------------------------------------------
// sums[b] = sum_e walked[e,b]
__global__ __launch_bounds__(256)
void k_reduce(const float* __restrict__ w, long long ws_e, long long ws_b,
              int E, float* __restrict__ sums) {
  __shared__ float part[256];
  const int lane_b = threadIdx.x & (BATCH - 1);
  long long row    = ((long long)blockIdx.x * blockDim.x + threadIdx.x) >> 4;
  long long stride = ((long long)gridDim.x * blockDim.x) >> 4;
  float acc = 0.0f;
  for (long long e = row; e < E; e += stride)
    acc += w[e * ws_e + (long long)lane_b * ws_b];
  part[threadIdx.x] = acc;
  __syncthreads();
  if (threadIdx.x < BATCH) {
    float s = 0.0f;
    for (int k = threadIdx.x; k < 256; k += BATCH) s += part[k];
    fatomic_add(&sums[threadIdx.x], s);
  }
}

// v = w[e,b]/(sums[b]+1e-6); optionally write back into w; always write the
// row-major output slice out_slice[b*os_b + e] (non-temporal: write-once).
__global__ void k_normalize(float* __restrict__ w, long long ws_e, long long ws_b,
                            float* __restrict__ out_slice, long long os_b,
                            const float* __restrict__ sums, int E, int write_w) {
  int b = blockIdx.y;
  float inv = 1.0f / (sums[b] + 1e-6f);
  for (long long e = (long long)blockIdx.x * blockDim.x + threadIdx.x; e < E;
       e += (long long)gridDim.x * blockDim.x) {
    long long wi = e * ws_e + (long long)b * ws_b;
    float v = w[wi] * inv;
    if (write_w) {
      w[wi] = v;
      __builtin_nontemporal_store(v, &out_slice[(long long)b * os_b + e]);
    } else {
      out_slice[(long long)b * os_b + e] = v;   // aliases w (fallback path)
    }
  }
}

// ---------------------------------------------------------------------------
extern "C" void kernel_launch(void* const* d_in, const int* in_sizes, int n_in,
                              void* d_out, int out_size, void* d_ws, size_t ws_size,
                              hipStream_t stream) {
  const float* rels_seq = (const float*)d_in[0];
  const float* init_ent = (const float*)d_in[1];
  const int*   heads    = (const int*)d_in[2];
  const int*   rels     = (const int*)d_in[3];
  const int*   tails    = (const int*)d_in[4];
  float*       out      = (float*)d_out;

  const int T = in_sizes[2];
  const int E = in_sizes[1] / BATCH;
  const int R = in_sizes[0] / (BATCH * HOPS);
  const long long outB = (long long)(HOPS + 1) * E;   // batch stride in d_out

  const int  threads = 256;
  const int  gE      = (E + threads - 1) / threads;   // blocks to cover E
  const dim3 gridBE((unsigned)gE, BATCH);

  // workspace partition (fast path): ent_t | walked_t | r_t | sums
  float* curT  = (float*)d_ws;
  float* nxtT  = curT + (long long)E * BATCH;
  float* rT    = nxtT + (long long)E * BATCH;
  float* sumsF = rT   + (long long)R * BATCH;
  const size_t need =
      ((size_t)E * BATCH * 2 + (size_t)R * BATCH + BATCH) * sizeof(float);
  const bool fast = ws_size >= need;

  // hop 0 output = init_ent
  k_copy_hop0<<<gridBE, threads, 0, stream>>>(init_ent, out, E, outB);

  if (fast) {
    k_transpose_in<<<4096, threads, 0, stream>>>(init_ent, curT, E);
    for (int hop = 0; hop < HOPS; ++hop) {
      k_zero<<<6400, threads, 0, stream>>>(nxtT, (long long)E * BATCH);
      k_zero<<<1, 32, 0, stream>>>(sumsF, BATCH);
      k_prep_r<<<(R * BATCH + threads - 1) / threads, threads, 0, stream>>>(
          rels_seq, rT, hop, R);
      k_scatter<<<SCATTER_BLOCKS, SCATTER_THREADS, 0, stream>>>(
          curT, BATCH, 1, rT, BATCH, 1, nxtT, BATCH, 1,
          heads, rels, tails, T);
      k_reduce<<<512, 256, 0, stream>>>(nxtT, BATCH, 1, E, sumsF);
      k_normalize<<<gridBE, threads, 0, stream>>>(
          nxtT, BATCH, 1, out + (long long)(hop + 1) * E, outB, sumsF, E, 1);
      float* t = curT; curT = nxtT; nxtT = t;   // ping-pong
    }
  } else {
    // degraded path: accumulate directly inside d_out slices, strided batch
    float* sums = (float*)d_ws;                 // needs only 64 bytes
    for (int hop = 0; hop < HOPS; ++hop) {
      float* w = out + (long long)(hop + 1) * E;          // [e*1 + b*outB]
      k_zero_slice<<<gridBE, threads, 0, stream>>>(w, outB, E);
      k_zero<<<1, 32, 0, stream>>>(sums, BATCH);
      const float* ent  = (hop == 0) ? init_ent : out + (long long)hop * E;
      long long    es_b = (hop == 0) ? (long long)E : outB;
      k_scatter<<<SCATTER_BLOCKS, SCATTER_THREADS, 0, stream>>>(
          ent, 1, es_b,
          rels_seq + (long long)hop * R, 1, (long long)HOPS * R,
          w, 1, outB, heads, rels, tails, T);
      k_reduce<<<512, 256, 0, stream>>>(w, 1, outB, E, sums);
      k_normalize<<<gridBE, threads, 0, stream>>>(w, 1, outB, w, outB, sums,
                                                  E, 0);
    }
  }
}